// SelfInteraction_46428596469937
// MI455X (gfx1250) — compile-verified
//
#include <hip/hip_runtime.h>

// fp32 WMMA fragment types for gfx1250
typedef float v2f __attribute__((ext_vector_type(2)));
typedef float v8f __attribute__((ext_vector_type(8)));

#define NPTS 4096
#define N_IN_ROWS 1024   // 64*(1+3+5+7)
#define N_OUT_ROWS 2048  // 128*(1+3+5+7)

// Equivariant block-diagonal linear: per (batch, l-block)
//   Out[128, D*4096] = W[128,64] x X[64, D*4096]
// Specialized per l-block so NCOLS is a compile-time constant -> all B loads
// and C stores use immediate offsets (max ~7.0 MB < signed 24-bit IOFFSET).
// One workgroup = 8 waves covering all 8 M-tiles of a 64-column panel, so the
// X panel is reused 8x through WGP$/L2.
template <int D, int ROW_IN, int ROW_OUT, int BLK>
__global__ __launch_bounds__(256) void SelfInteraction_wmma_kernel(
    const float* __restrict__ x, const float* __restrict__ w,
    float* __restrict__ out) {
  constexpr int NCOLS = D * NPTS;  // N of this block's GEMM (row stride)

  const int tid    = threadIdx.x;
  const int wave   = tid >> 5;          // 0..7  -> M tile
  const int lane   = tid & 31;
  const int lane16 = lane & 15;
  const int khalf  = (lane >> 4) << 1;  // 0 or 2 (upper half-wave: K+2/K+3)

  const int nb     = blockIdx.y;        // batch
  const int n_base = blockIdx.x * 64;   // first column of this WG's panel

  const float* xb = x   + (size_t)nb * N_IN_ROWS  * NPTS + (size_t)ROW_IN  * NPTS;
  float*       ob = out + (size_t)nb * N_OUT_ROWS * NPTS + (size_t)ROW_OUT * NPTS;

  // ---- A fragments: W[16x64] tile for this wave, 16 k-steps of 16x4 ----
  // A layout (16x4 f32): lane L<16 holds {A[L,k+0], A[L,k+1]},
  //                      lane L+16 holds {A[L,k+2], A[L,k+3]}  -> v2f each
  const int m = wave * 16 + lane16;
  const float* wp = w + BLK * 8192 + m * 64 + khalf;
  v2f a[16];
#pragma unroll
  for (int kk = 0; kk < 16; ++kk) {
    a[kk] = *(const v2f*)(wp + 4 * kk);  // contiguous pair, 8B aligned
  }

  // ---- 4 N-tiles of 16 columns each ----
#pragma unroll
  for (int t = 0; t < 4; ++t) {
    const int n = n_base + t * 16 + lane16;
    // B layout (4x16 f32): VGPR0 = rows {k, k+2}, VGPR1 = rows {k+1, k+3},
    // columns striped over lanes -> same khalf split as A.
    const float* xp = xb + khalf * NCOLS + n;  // single base; imm offsets below

    v8f c = {};
#pragma unroll
    for (int kk = 0; kk < 16; ++kk) {
      v2f bf;
      bf.x = xp[kk * 4 * NCOLS];          // row k+khalf   (imm offset)
      bf.y = xp[kk * 4 * NCOLS + NCOLS];  // row k+khalf+1 (imm offset)
      c = __builtin_amdgcn_wmma_f32_16x16x4_f32(
          /*neg_a=*/false, a[kk], /*neg_b=*/false, bf,
          /*c_mod=*/(short)0, c, /*reuse_a=*/false, /*reuse_b=*/false);
    }

    // C/D layout: VGPR v -> row (m0 + v) for lanes 0-15, (m0 + v + 8) for 16-31
    // Output is streamed (never re-read): nontemporal stores keep the shared
    // X panels resident in WGP$/L2.
    float* op = ob + (size_t)(wave * 16 + (lane >> 4) * 8) * NCOLS + n;
#pragma unroll
    for (int v = 0; v < 8; ++v) {
      __builtin_nontemporal_store(c[v], op + v * NCOLS);
    }
  }
}

extern "C" void kernel_launch(void* const* d_in, const int* in_sizes, int n_in,
                              void* d_out, int out_size, void* d_ws, size_t ws_size,
                              hipStream_t stream) {
  const float* x = (const float*)d_in[0];   // [8, 1024, 4096] f32
  const float* w = (const float*)d_in[1];   // [32768] f32
  float* out = (float*)d_out;               // [8, 2048, 4096] f32

  dim3 block(256);
  // grid.x = (D*4096)/64 column panels, grid.y = 8 batches
  hipLaunchKernelGGL((SelfInteraction_wmma_kernel<1, 0,   0,    0>),
                     dim3(64, 8),  block, 0, stream, x, w, out);
  hipLaunchKernelGGL((SelfInteraction_wmma_kernel<3, 64,  128,  1>),
                     dim3(192, 8), block, 0, stream, x, w, out);
  hipLaunchKernelGGL((SelfInteraction_wmma_kernel<5, 256, 512,  2>),
                     dim3(320, 8), block, 0, stream, x, w, out);
  hipLaunchKernelGGL((SelfInteraction_wmma_kernel<7, 576, 1152, 3>),
                     dim3(448, 8), block, 0, stream, x, w, out);
}